// FastRNNDetector_1434519076866
// MI455X (gfx1250) — compile-verified
//
#include <hip/hip_runtime.h>
#include <hip/hip_bf16.h>
#include <math.h>

// ---------------- problem constants (match reference) ----------------
#define B_N   8
#define K_N   80
#define HF_N  160
#define WF_N  160
#define S_N   (HF_N * WF_N)     // 25600 locations per image
#define M_N   64                // GT boxes per image
#define STRIDE_F 8.0f           // 1280 / 160
#define ALPHA_F  0.25f
#define CLIP_MAX 1279.0f        // IMG_W-1 == IMG_H-1

#define THREADS 256
#define NBLOCKS ((B_N * S_N) / THREADS)   // 800, exact
#define KBATCH  8                          // loads in flight per thread

typedef float v2f __attribute__((ext_vector_type(2)));
typedef float v8f __attribute__((ext_vector_type(8)));

// ---------------- wave32 sum via V_WMMA_F32_16X16X4_F32 ----------------
// A is 16x4 striped: lanes 0-15 supply {K0,K1}, lanes 16-31 supply {K2,K3},
// row M = lane%16. Per-lane partial goes in a[0] (A[m,0]=v_m, A[m,2]=v_{m+16}),
// a[1]=0. With B = ones(4x16): D[m,n] = v_m + v_{m+16} replicated over n.
// C/D layout: VGPR j, lanes 0-15 hold M=j, lanes 16-31 hold M=8+j; summing the
// 8 C vgprs gives half-sums, shfl_xor(16) completes the wave32 sum.
// Deterministic order; EXEC is all-1s at every call site.
__device__ __forceinline__ float wave_sum32(float v) {
#if __has_builtin(__builtin_amdgcn_wmma_f32_16x16x4_f32)
    v2f a; a[0] = v;    a[1] = 0.0f;
    v2f b; b[0] = 1.0f; b[1] = 1.0f;
    v8f c = {};
    c = __builtin_amdgcn_wmma_f32_16x16x4_f32(
            /*neg_a=*/false, a, /*neg_b=*/false, b,
            /*c_mod=*/(short)0, c, /*reuse_a=*/false, /*reuse_b=*/false);
    float r = ((c[0] + c[1]) + (c[2] + c[3])) + ((c[4] + c[5]) + (c[6] + c[7]));
    r += __shfl_xor(r, 16, 32);
    return r;
#else
    for (int off = 16; off > 0; off >>= 1) v += __shfl_xor(v, off, 32);
    return v;
#endif
}

// ---------------- fused assignment + focal/reg/ctr loss ----------------
__global__ __launch_bounds__(THREADS) void fcos_loss_kernel(
    const float* __restrict__ cls,   // (B,K,H,W)
    const float* __restrict__ reg,   // (B,4,H,W)
    const float* __restrict__ ctr,   // (B,1,H,W)
    const float* __restrict__ gtb,   // (B,M,4) xyxy
    const int*   __restrict__ gtl,   // (B,M)
    float* __restrict__ partials)    // (gridDim.x, 4): cls,reg,ctr,pos
{
    const int tid  = threadIdx.x;
    const int bpb  = S_N / THREADS;            // 100 blocks per batch image
    const int b    = blockIdx.x / bpb;
    const int s    = (blockIdx.x % bpb) * THREADS + tid;

    // Stage clipped boxes + labels in LDS (256 floats == M*4, one per thread)
    __shared__ __align__(16) float sbox[M_N * 4];
    __shared__ int slab[M_N];
    {
        float v = gtb[(size_t)b * M_N * 4 + tid];
        v = fminf(fmaxf(v, 0.0f), CLIP_MAX);
        sbox[tid] = v;
        if (tid < M_N) slab[tid] = gtl[b * M_N + tid];
    }
    __syncthreads();

    const int   h  = s / WF_N;
    const int   w  = s - h * WF_N;
    const float px = ((float)w + 0.5f) * STRIDE_F;
    const float py = ((float)h + 0.5f) * STRIDE_F;

    // ---- FCOS assignment: min-area inside box (first-occurrence tie-break)
    const float4* __restrict__ sb4 = (const float4*)sbox;   // ds_load_b128 per box
    float bestArea = __builtin_inff();
    int   best     = 0;
#pragma unroll 4
    for (int m = 0; m < M_N; ++m) {
        const float4 bx = sb4[m];
        const float l = px - bx.x, t = py - bx.y, r = bx.z - px, bo = bx.w - py;
        const bool inside = (l > 0.0f) && (t > 0.0f) && (r > 0.0f) && (bo > 0.0f);
        const float area = fmaxf(bx.z - bx.x, 0.0f) * fmaxf(bx.w - bx.y, 0.0f);
        const float a = inside ? area : __builtin_inff();
        if (a < bestArea) { bestArea = a; best = m; }
    }

    const float4 bb = sb4[best];
    const float l  = px - bb.x, t = py - bb.y, r = bb.z - px, bo = bb.w - py;
    const bool  pos  = (l > 0.0f) && (t > 0.0f) && (r > 0.0f) && (bo > 0.0f);
    const float posf = pos ? 1.0f : 0.0f;
    const int   labm1 = (pos ? slab[best] : 0) - 1;   // -1 => no positive class

    // ---- focal classification loss: strided pass over K logits,
    //      batched 8-deep so 8 loads are in flight before compute.
    const float* __restrict__ cp = cls + (size_t)b * K_N * S_N + s;
    float sum_cls = 0.0f;
    for (int k0 = 0; k0 < K_N; k0 += KBATCH) {
        float xv[KBATCH];
#pragma unroll
        for (int i = 0; i < KBATCH; ++i) {
            xv[i] = cp[(size_t)(k0 + i) * S_N];
            if (k0 + i + KBATCH < K_N)       // prefetch next batch's rows
                __builtin_prefetch(cp + (size_t)(k0 + i + KBATCH) * S_N, 0, 1);
        }
#pragma unroll
        for (int i = 0; i < KBATCH; ++i) {
            const float x   = xv[i];
            const float tgt = ((k0 + i) == labm1) ? 1.0f : 0.0f;
            const float e   = expf(-fabsf(x));      // single exp, shared
            const float L   = log1pf(e);
            const float inv = 1.0f / (1.0f + e);
            const float p   = (x >= 0.0f) ? inv : e * inv;   // sigmoid(x)
            const float ce  = fmaxf(x, 0.0f) - x * tgt + L;
            const float pt  = p * tgt + (1.0f - p) * (1.0f - tgt);
            const float at  = ALPHA_F * tgt + (1.0f - ALPHA_F) * (1.0f - tgt);
            const float om  = 1.0f - pt;
            sum_cls += at * ce * om * om;
        }
    }

    // ---- L1 regression loss on positives (targets in stride units)
    const float lt4[4] = { l / STRIDE_F, t / STRIDE_F, r / STRIDE_F, bo / STRIDE_F };
    float sum_reg = 0.0f;
#pragma unroll
    for (int c = 0; c < 4; ++c) {
        const float rv = reg[((size_t)b * 4 + c) * S_N + s];
        sum_reg += fabsf(rv - lt4[c]);
    }
    sum_reg *= posf;

    // ---- centerness BCE; clip must PROPAGATE NaN exactly like jnp.clip
    const float lrmin = fminf(l, r);
    const float lrmax = fmaxf(fmaxf(l, r), 1e-6f);
    const float tbmin = fminf(t, bo);
    const float tbmax = fmaxf(fmaxf(t, bo), 1e-6f);
    float ct = sqrtf((lrmin / lrmax) * (tbmin / tbmax));   // NaN possible: keep it
    ct = (ct < 0.0f) ? 0.0f : ((ct > 1.0f) ? 1.0f : ct);   // NaN falls through
    const float xc = ctr[(size_t)b * S_N + s];
    const float sum_ctr =
        (fmaxf(xc, 0.0f) - xc * ct + log1pf(expf(-fabsf(xc)))) * posf; // NaN*0=NaN (ref)

    // ---- deterministic block reduction: WMMA wave sums -> LDS -> thread 0
    const float rc = wave_sum32(sum_cls);
    const float rr = wave_sum32(sum_reg);
    const float rt = wave_sum32(sum_ctr);
    const float rp = wave_sum32(posf);

    __shared__ float wsum[THREADS / 32][4];
    const int lane = tid & 31, wv = tid >> 5;
    if (lane == 0) { wsum[wv][0] = rc; wsum[wv][1] = rr; wsum[wv][2] = rt; wsum[wv][3] = rp; }
    __syncthreads();
    if (tid == 0) {
        float a0 = 0.f, a1 = 0.f, a2 = 0.f, a3 = 0.f;
#pragma unroll
        for (int i = 0; i < THREADS / 32; ++i) {
            a0 += wsum[i][0]; a1 += wsum[i][1]; a2 += wsum[i][2]; a3 += wsum[i][3];
        }
        float* pp = partials + (size_t)blockIdx.x * 4;
        pp[0] = a0; pp[1] = a1; pp[2] = a2; pp[3] = a3;
    }
}

// ---------------- deterministic final reduction + normalization ----------------
__global__ __launch_bounds__(THREADS) void fcos_finalize_kernel(
    const float* __restrict__ partials, int nblocks, float* __restrict__ out)
{
    __shared__ float s0[THREADS], s1[THREADS], s2[THREADS], s3[THREADS];
    const int t = threadIdx.x;
    float a0 = 0.f, a1 = 0.f, a2 = 0.f, a3 = 0.f;
    for (int i = t; i < nblocks; i += THREADS) {
        const float* pp = partials + (size_t)i * 4;
        a0 += pp[0]; a1 += pp[1]; a2 += pp[2]; a3 += pp[3];
    }
    s0[t] = a0; s1[t] = a1; s2[t] = a2; s3[t] = a3;
    __syncthreads();
    for (int off = THREADS / 2; off > 0; off >>= 1) {
        if (t < off) {
            s0[t] += s0[t + off]; s1[t] += s1[t + off];
            s2[t] += s2[t + off]; s3[t] += s3[t + off];
        }
        __syncthreads();
    }
    if (t == 0) {
        const float np = fmaxf(s3[0], 1.0f);
        out[0] = s0[0] / np;   // loss_cls
        out[1] = s1[0] / np;   // loss_reg
        out[2] = s2[0] / np;   // loss_ctr
    }
}

extern "C" void kernel_launch(void* const* d_in, const int* in_sizes, int n_in,
                              void* d_out, int out_size, void* d_ws, size_t ws_size,
                              hipStream_t stream) {
    (void)in_sizes; (void)n_in; (void)out_size; (void)ws_size;
    const float* cls = (const float*)d_in[0];   // (B,K,H,W) f32
    const float* reg = (const float*)d_in[1];   // (B,4,H,W) f32
    const float* ctr = (const float*)d_in[2];   // (B,1,H,W) f32
    const float* gtb = (const float*)d_in[3];   // (B,M,4)   f32
    const int*   gtl = (const int*)d_in[4];     // (B,M)     i32
    float*       out = (float*)d_out;           // 3 floats
    float*       partials = (float*)d_ws;       // NBLOCKS*4 floats (12.8 KB)

    fcos_loss_kernel<<<NBLOCKS, THREADS, 0, stream>>>(cls, reg, ctr, gtb, gtl, partials);
    fcos_finalize_kernel<<<1, THREADS, 0, stream>>>(partials, NBLOCKS, out);
}